// LSTMAutoencoder_16020228014585
// MI455X (gfx1250) — compile-verified
//
#include <hip/hip_runtime.h>

// ---------------------------------------------------------------------------
// LSTM autoencoder for MI455X (gfx1250) — pipelined persistent kernel.
//   B=64, T=1024, layers: (256->256)(256->128)(128->256)(256->256)
//   All 4 layers resident at once (56 WGs); layer l step t starts as soon as
//   layer l-1 step t is published (monotonic counters as producer flags),
//   collapsing the critical path from 4096 to ~1027 sequential steps.
//   Per step: z = [x_t | h_{t-1}] @ [W;U] + b via V_WMMA_F32_16X16X4_F32,
//   weights pre-swizzled into LDS as ready-made B-fragments, h double-buffered
//   in global memory with one device-scope atomic barrier per step.
// ---------------------------------------------------------------------------

typedef __attribute__((ext_vector_type(2))) float v2f;
typedef __attribute__((ext_vector_type(8))) float v8f;

#define TSTEPS 1024
#define BATCH  64

__device__ __forceinline__ float sigmoidf_(float x) {
    return 1.0f / (1.0f + __expf(-x));
}

// Zero h double-buffers and step counters for all 4 layers (runs every call,
// so graph replays are deterministic).
__global__ void init_ws_kernel(float* __restrict__ hbufs, unsigned* __restrict__ ctrs) {
    const int stride = gridDim.x * blockDim.x;
    int i = blockIdx.x * blockDim.x + threadIdx.x;
    const int nh = 4 * 2 * BATCH * 256;      // 4 layers x 2 buffers x 64x256
    for (int k = i; k < nh; k += stride) hbufs[k] = 0.0f;
    if (i < 4) ctrs[i * 16] = 0u;
}

template <int F, int H>
__device__ __forceinline__ void lstm_layer_dev(
    const int wgid,                     // 0..NWG-1 within this layer
    const float* __restrict__ xin,      // [B,T,F]  (previous layer's output)
    const float* __restrict__ W,        // [F,4H]
    const float* __restrict__ U,        // [H,4H]
    const float* __restrict__ bias,     // [4H]
    float* __restrict__ out,            // [B,T,H]
    float* __restrict__ hbuf,           // [2][B*H], buf0 zeroed
    unsigned* __restrict__ ctr,         // this layer's step counter (zeroed)
    const unsigned* __restrict__ ctrPrev, // producer counter (nullptr for L1)
    const unsigned nwgPrev)             // producer's workgroup count
{
    constexpr int K   = F + H;      // fused GEMM K dim
    constexpr int G   = 4 * H;      // gate width
    constexpr int NWG = H / 16;     // workgroups in this layer

    extern __shared__ float smem[];
    float* wlds = smem;                       // K*64 floats: swizzled B-fragments
    float* zbuf = smem + K * 64;              // 64 x 64 local z tile
    float* cst  = zbuf + 64 * 64;             // 64 x 16 cell state
    float* blds = cst + 1024;                 // 64 bias values

    const int tid = threadIdx.x;              // 256 threads = 8 waves (wave32)
    const int j0  = wgid * 16;                // first hidden unit owned by this WG

    // ---- one-time: swizzle [W;U] columns for this WG into B-fragment layout
    v2f* wv2 = (v2f*)wlds;
    for (int idx = tid; idx < K * 32; idx += 256) {
        const int l  = idx & 31;
        const int kt = idx >> 5;         // ks*4 + nt
        const int nt = kt & 3;
        const int ks = kt >> 2;
        const int hi = l >> 4;
        const int gcol = nt * H + j0 + (l & 15);
        const int k0 = 4 * ks + 2 * hi;  // even; F even => k0 and k0+1 same matrix
        const float* src = (k0 < F) ? (W + (size_t)k0 * G + gcol)
                                    : (U + (size_t)(k0 - F) * G + gcol);
        v2f w;
        w.x = src[0];
        w.y = src[G];
        wv2[idx] = w;
    }
    for (int i = tid; i < 1024; i += 256) cst[i] = 0.0f;
    if (tid < 64) blds[tid] = bias[(tid >> 4) * H + j0 + (tid & 15)];
    __syncthreads();

    // ---- wave tiling: wave w -> N-tile nt=w%4 (== gate), M-tiles {w/4, w/4+2}
    const int lane  = tid & 31;
    const int wid   = tid >> 5;
    const int nt    = wid & 3;
    const int mt0   = wid >> 2;
    const int hi    = lane >> 4;
    const int m16   = lane & 15;
    const int rowA0 = mt0 * 16 + m16;          // batch rows for A fragments
    const int rowA1 = rowA0 + 32;
    const int colz  = nt * 16 + m16;           // z store coords
    const int rowz0 = mt0 * 16 + hi * 8;
    const int rowz1 = rowz0 + 32;

    const v2f* wfrag = (const v2f*)wlds + nt * 32 + lane;   // stride 128 per ks

    // pipeline fill: wait for producer's step 0 before first read of xin
    if (ctrPrev != nullptr) {
        if (tid == 0) {
            while (__hip_atomic_load(ctrPrev, __ATOMIC_ACQUIRE, __HIP_MEMORY_SCOPE_AGENT)
                   < nwgPrev) {
                __builtin_amdgcn_s_sleep(2);
            }
        }
        __syncthreads();
        __threadfence();
    }

    for (int t = 0; t < TSTEPS; ++t) {
        const float* xr0 = xin + ((size_t)rowA0 * TSTEPS + t) * F + 2 * hi;
        const float* xr1 = xin + ((size_t)rowA1 * TSTEPS + t) * F + 2 * hi;
        const float* hr  = hbuf + (size_t)(t & 1) * (BATCH * H);   // h_{t-1}
        const float* hr0 = hr + rowA0 * H + 2 * hi;
        const float* hr1 = hr + rowA1 * H + 2 * hi;

        v8f acc0 = {0.f, 0.f, 0.f, 0.f, 0.f, 0.f, 0.f, 0.f};
        v8f acc1 = {0.f, 0.f, 0.f, 0.f, 0.f, 0.f, 0.f, 0.f};

        // z += x_t @ W
        #pragma unroll 4
        for (int ks = 0; ks < F / 4; ++ks) {
            v2f a0 = *(const v2f*)(xr0 + 4 * ks);
            v2f a1 = *(const v2f*)(xr1 + 4 * ks);
            v2f bf = wfrag[ks * 128];
            acc0 = __builtin_amdgcn_wmma_f32_16x16x4_f32(false, a0, false, bf,
                                                         (short)0, acc0, false, false);
            acc1 = __builtin_amdgcn_wmma_f32_16x16x4_f32(false, a1, false, bf,
                                                         (short)0, acc1, false, false);
        }
        // z += h_{t-1} @ U
        #pragma unroll 4
        for (int ks = 0; ks < H / 4; ++ks) {
            v2f a0 = *(const v2f*)(hr0 + 4 * ks);
            v2f a1 = *(const v2f*)(hr1 + 4 * ks);
            v2f bf = wfrag[(F / 4 + ks) * 128];
            acc0 = __builtin_amdgcn_wmma_f32_16x16x4_f32(false, a0, false, bf,
                                                         (short)0, acc0, false, false);
            acc1 = __builtin_amdgcn_wmma_f32_16x16x4_f32(false, a1, false, bf,
                                                         (short)0, acc1, false, false);
        }

        // spill z tiles to LDS (C/D layout: VGPR r, lane l -> row r+8*(l/16))
        #pragma unroll
        for (int r = 0; r < 8; ++r) {
            zbuf[(rowz0 + r) * 64 + colz] = acc0[r];
            zbuf[(rowz1 + r) * 64 + colz] = acc1[r];
        }
        __syncthreads();

        // gates + state update for this WG's 64x16 slice
        float* hw = hbuf + (size_t)((t + 1) & 1) * (BATCH * H);
        #pragma unroll
        for (int q = 0; q < 4; ++q) {
            const int idx = tid + q * 256;    // 0..1023
            const int m = idx >> 4;
            const int j = idx & 15;
            const float zi = zbuf[m * 64 +  0 + j] + blds[ 0 + j];
            const float zf = zbuf[m * 64 + 16 + j] + blds[16 + j];
            const float zg = zbuf[m * 64 + 32 + j] + blds[32 + j];
            const float zo = zbuf[m * 64 + 48 + j] + blds[48 + j];
            const float ig = sigmoidf_(zi);
            const float fg = sigmoidf_(zf);
            const float og = sigmoidf_(zo);
            const float gg = fmaxf(zg, 0.0f);
            const float c  = fg * cst[idx] + ig * gg;
            cst[idx] = c;
            const float h = og * fmaxf(c, 0.0f);
            hw[m * H + j0 + j] = h;
            out[((size_t)m * TSTEPS + t) * H + j0 + j] = h;
        }

        // device-scope barrier: publish our step, wait for layer peers, and
        // wait for the producer layer to publish the next timestep's input.
        __threadfence();                       // release h/out stores
        __syncthreads();
        if (tid == 0) {
            __hip_atomic_fetch_add(ctr, 1u, __ATOMIC_ACQ_REL, __HIP_MEMORY_SCOPE_AGENT);
            const unsigned target = (unsigned)(t + 1) * (unsigned)NWG;
            while (__hip_atomic_load(ctr, __ATOMIC_ACQUIRE, __HIP_MEMORY_SCOPE_AGENT) < target) {
                __builtin_amdgcn_s_sleep(2);
            }
            if (ctrPrev != nullptr && (t + 1) < TSTEPS) {
                const unsigned ptarget = (unsigned)(t + 2) * nwgPrev;
                while (__hip_atomic_load(ctrPrev, __ATOMIC_ACQUIRE,
                                         __HIP_MEMORY_SCOPE_AGENT) < ptarget) {
                    __builtin_amdgcn_s_sleep(2);
                }
            }
        }
        __syncthreads();
        __threadfence();                       // acquire: drop stale cached data
    }
}

struct LstmArgs {
    const float *x;
    const float *W1, *U1, *b1, *W2, *U2, *b2, *W3, *U3, *b3, *W4, *U4, *b4;
    float *actA, *actB, *actC, *out;
    float *hbufs;
    unsigned *ctrs;
};

// 56 workgroups: L1 -> [0,16), L2 -> [16,24), L3 -> [24,40), L4 -> [40,56)
__global__ void lstm_pipeline_kernel(LstmArgs a) {
    const int wg = blockIdx.x;
    float* hb0 = a.hbufs + (size_t)0 * 2 * BATCH * 256;
    float* hb1 = a.hbufs + (size_t)1 * 2 * BATCH * 256;
    float* hb2 = a.hbufs + (size_t)2 * 2 * BATCH * 256;
    float* hb3 = a.hbufs + (size_t)3 * 2 * BATCH * 256;
    unsigned* c0 = a.ctrs + 0 * 16;
    unsigned* c1 = a.ctrs + 1 * 16;
    unsigned* c2 = a.ctrs + 2 * 16;
    unsigned* c3 = a.ctrs + 3 * 16;

    if (wg < 16) {
        lstm_layer_dev<256, 256>(wg,      a.x,    a.W1, a.U1, a.b1, a.actA, hb0, c0,
                                 nullptr, 0u);
    } else if (wg < 24) {
        lstm_layer_dev<256, 128>(wg - 16, a.actA, a.W2, a.U2, a.b2, a.actB, hb1, c1,
                                 c0, 16u);
    } else if (wg < 40) {
        lstm_layer_dev<128, 256>(wg - 24, a.actB, a.W3, a.U3, a.b3, a.actC, hb2, c2,
                                 c1, 8u);
    } else {
        lstm_layer_dev<256, 256>(wg - 40, a.actC, a.W4, a.U4, a.b4, a.out,  hb3, c3,
                                 c2, 16u);
    }
}

// ---------------------------------------------------------------------------
extern "C" void kernel_launch(void* const* d_in, const int* in_sizes, int n_in,
                              void* d_out, int out_size, void* d_ws, size_t ws_size,
                              hipStream_t stream) {
    (void)in_sizes; (void)n_in; (void)out_size; (void)ws_size;

    LstmArgs a;
    a.x  = (const float*)d_in[0];
    a.W1 = (const float*)d_in[1];
    a.U1 = (const float*)d_in[2];
    a.b1 = (const float*)d_in[3];
    a.W2 = (const float*)d_in[4];
    a.U2 = (const float*)d_in[5];
    a.b2 = (const float*)d_in[6];
    a.W3 = (const float*)d_in[7];
    a.U3 = (const float*)d_in[8];
    a.b3 = (const float*)d_in[9];
    a.W4 = (const float*)d_in[10];
    a.U4 = (const float*)d_in[11];
    a.b4 = (const float*)d_in[12];

    // workspace layout (concurrent layers -> distinct activation buffers)
    char* ws = (char*)d_ws;
    a.actA  = (float*)ws;                                              // [B,T,256] 64MB
    a.actB  = (float*)(ws + (size_t)BATCH * TSTEPS * 256 * 4);         // [B,T,128] 32MB
    a.actC  = (float*)(ws + (size_t)BATCH * TSTEPS * (256 + 128) * 4); // [B,T,256] 64MB
    a.hbufs = (float*)(ws + (size_t)BATCH * TSTEPS * (256 + 128 + 256) * 4);
    a.ctrs  = (unsigned*)((char*)a.hbufs + (size_t)4 * 2 * BATCH * 256 * 4);
    a.out   = (float*)d_out;

    const int sh512 = 4 * (512 * 64 + 64 * 64 + 1024 + 64);   // ~149 KB LDS (max layer)
    (void)hipFuncSetAttribute(reinterpret_cast<const void*>(&lstm_pipeline_kernel),
                              hipFuncAttributeMaxDynamicSharedMemorySize, sh512);

    init_ws_kernel<<<64, 256, 0, stream>>>(a.hbufs, a.ctrs);
    lstm_pipeline_kernel<<<56, 256, sh512, stream>>>(a);
}